// NTMMemory_16509854286496
// MI455X (gfx1250) — compile-verified
//
#include <hip/hip_runtime.h>

// NTM memory update, fused single-pass streaming kernel for gfx1250 (MI455X).
// read_out = einsum('bn,bnm->bm', rw, mem)   (folded into the stream via WMMA)
// new_mem  = mem*(1 - w*e) + w*a             (elementwise, non-temporal stream)

typedef __attribute__((ext_vector_type(2))) float v2f;
typedef __attribute__((ext_vector_type(8))) float v8f;

constexpr int B_ = 64;
constexpr int N_ = 1024;
constexpr int M_ = 512;
constexpr int NCH = 8;                       // chunks along N
constexpr int ROWS = N_ / NCH;               // 128 rows per chunk
constexpr int COLG = 4;                      // column groups of 128 (8 waves * 16 cols)
constexpr int THREADS = 256;                 // 8 wave32 per block

__global__ __launch_bounds__(THREADS)
void ntm_fused_kernel(const float* __restrict__ memory,
                      const float* __restrict__ read_w,
                      const float* __restrict__ write_w,
                      const float* __restrict__ erase_v,
                      const float* __restrict__ add_v,
                      float* __restrict__ new_memory,   // d_out + B*M
                      float* __restrict__ partial)      // [B][NCH][M] in d_ws
{
    __shared__ float sRW[ROWS];
    __shared__ float sWW[ROWS];

    const int id     = blockIdx.x;
    const int nChunk = id & (NCH - 1);
    const int colG   = (id >> 3) & (COLG - 1);
    const int b      = id >> 5;

    const int nBase = nChunk * ROWS;

    // Stage this chunk's read/write weights in LDS (broadcast-friendly).
    const int t = threadIdx.x;
    if (t < ROWS) sRW[t] = read_w[b * N_ + nBase + t];
    else          sWW[t - ROWS] = write_w[b * N_ + nBase + (t - ROWS)];
    __syncthreads();

    const int wave = t >> 5;
    const int lane = t & 31;
    const int g    = lane >> 4;   // half-wave group: selects K=0,1 vs K=2,3
    const int col  = lane & 15;
    const int m    = colG * 128 + wave * 16 + col;

    const float em = erase_v[b * M_ + m];
    const float am = add_v[b * M_ + m];

    const float* memB = memory     + (size_t)b * N_ * M_;
    float*       nmB  = new_memory + (size_t)b * N_ * M_;

    // D accumulator: 16x16 f32, all rows identical by construction of A.
    v8f c = {0.f, 0.f, 0.f, 0.f, 0.f, 0.f, 0.f, 0.f};

    #pragma unroll 4
    for (int k = 0; k < ROWS; k += 4) {
        const int r   = k + 2 * g;          // local row for this half-wave
        const int row = nBase + r;

        // A-matrix (16x4 f32 layout): lanes 0-15 -> K=0,1 ; lanes 16-31 -> K=2,3.
        // Replicating rw across all 16 M-rows makes every D row the GEMV partial.
        const float rw0 = sRW[r];
        const float rw1 = sRW[r + 1];
        const float w0  = sWW[r];
        const float w1  = sWW[r + 1];

        // B-matrix (4x16 f32 layout): VGPR0 = rows K=0/K=2, VGPR1 = rows K=1/K=3,
        // column = lane&15. Two coalesced b32 loads (64B per half-wave per row).
        const size_t off0 = (size_t)row * M_ + m;
        const float x0 = __builtin_nontemporal_load(&memB[off0]);
        const float x1 = __builtin_nontemporal_load(&memB[off0 + M_]);

        // Elementwise erase/add update, streamed out non-temporally.
        __builtin_nontemporal_store(fmaf(x0, fmaf(-w0, em, 1.0f), w0 * am), &nmB[off0]);
        __builtin_nontemporal_store(fmaf(x1, fmaf(-w1, em, 1.0f), w1 * am), &nmB[off0 + M_]);

        v2f Av = {rw0, rw1};
        v2f Bv = {x0, x1};
        // D = A x B + C  (V_WMMA_F32_16X16X4_F32)
        c = __builtin_amdgcn_wmma_f32_16x16x4_f32(false, Av, false, Bv,
                                                  (short)0, c, false, false);
    }

    // D row 0 (VGPR0, lanes 0-15) holds the partial read_out for our 16 columns.
    if (lane < 16) {
        partial[((size_t)b * NCH + nChunk) * M_ + m] = c[0];
    }
}

__global__ __launch_bounds__(THREADS)
void ntm_reduce_kernel(const float* __restrict__ partial,
                       float* __restrict__ read_out)   // d_out[0 .. B*M)
{
    const int idx = blockIdx.x * THREADS + threadIdx.x;   // < B*M
    const int b = idx >> 9;          // / 512
    const int m = idx & (M_ - 1);
    float s = 0.f;
    #pragma unroll
    for (int ch = 0; ch < NCH; ++ch)
        s += partial[((size_t)b * NCH + ch) * M_ + m];
    read_out[idx] = s;
}

extern "C" void kernel_launch(void* const* d_in, const int* in_sizes, int n_in,
                              void* d_out, int out_size, void* d_ws, size_t ws_size,
                              hipStream_t stream) {
    const float* memory  = (const float*)d_in[0];
    const float* read_w  = (const float*)d_in[1];
    const float* write_w = (const float*)d_in[2];
    const float* erase_v = (const float*)d_in[3];
    const float* add_v   = (const float*)d_in[4];

    float* out        = (float*)d_out;
    float* new_memory = out + (size_t)B_ * M_;   // tuple order: (read_out, new_memory)
    float* partial    = (float*)d_ws;            // B*NCH*M floats = 1 MB

    ntm_fused_kernel<<<B_ * COLG * NCH, THREADS, 0, stream>>>(
        memory, read_w, write_w, erase_v, add_v, new_memory, partial);

    ntm_reduce_kernel<<<(B_ * M_) / THREADS, THREADS, 0, stream>>>(partial, out);
}